// GraphConv_NCN_5592047419467
// MI455X (gfx1250) — compile-verified
//
#include <hip/hip_runtime.h>
#include <hip/hip_bf16.h>

typedef __attribute__((ext_vector_type(2))) float v2f;
typedef __attribute__((ext_vector_type(8))) float v8f;

#define D_FEAT 256

// ---------------------------------------------------------------------------
// Kernel 1: zero agg[N*256] (as float4) and deg/norm[N]
// ---------------------------------------------------------------------------
__global__ void zero_kernel(float4* __restrict__ agg4, float* __restrict__ nd,
                            int n4, int n_nodes) {
    int gid = blockIdx.x * blockDim.x + threadIdx.x;
    if (gid < n4) agg4[gid] = make_float4(0.f, 0.f, 0.f, 0.f);
    if (gid < n_nodes) nd[gid] = 0.f;
}

// ---------------------------------------------------------------------------
// Kernel 2: degree count (float accumulation; exact for deg < 2^24)
// ---------------------------------------------------------------------------
__global__ void deg_kernel(const int* __restrict__ rows, float* __restrict__ degf,
                           int n_edges) {
    int e = blockIdx.x * blockDim.x + threadIdx.x;
    if (e < n_edges) atomicAdd(&degf[rows[e]], 1.0f);
}

// ---------------------------------------------------------------------------
// Kernel 3: norm = rsqrt(1 + deg), in place
// ---------------------------------------------------------------------------
__global__ void norm_kernel(float* __restrict__ nd, int n_nodes) {
    int i = blockIdx.x * blockDim.x + threadIdx.x;
    if (i < n_nodes) nd[i] = rsqrtf(1.0f + nd[i]);
}

// ---------------------------------------------------------------------------
// Kernel 4: h_scaled = norm[i] * (x @ W^T + bias), via V_WMMA_F32_16X16X4_F32.
// Block = 256 threads = 8 waves. Each block handles a 16-row strip of x;
// wave w computes output column tiles [32w, 32w+16) and [32w+16, 32w+32).
//
// A (16x4 f32, MxK): lanes 0-15 hold M=lane (K=k,k+1 in v[0],v[1]);
//                    lanes 16-31 hold M=lane-16 (K=k+2,k+3).
// B (4x16 f32, KxN): lane n=L&15, K pair = k + 2*(L>>4); consecutive K in v[0],v[1].
//   B[k][n] = W[col0+n][k]  (out = x @ W^T, W stored [out][in] row-major)
// C/D (16x16 f32): VGPR r -> row r (lanes 0-15) / row r+8 (lanes 16-31), col = L&15.
// ---------------------------------------------------------------------------
__global__ __launch_bounds__(256) void gemm_norm_kernel(
    const float* __restrict__ x, const float* __restrict__ W,
    const float* __restrict__ bias, const float* __restrict__ norm,
    float* __restrict__ h, int n_rows) {
    const int wave = threadIdx.x >> 5;   // 0..7
    const int lane = threadIdx.x & 31;
    const int row0 = blockIdx.x * 16;    // wave-uniform
    if (row0 >= n_rows) return;          // whole wave exits together (EXEC stays full)

    const int col0  = wave * 32;         // two 16-wide column tiles per wave
    const int m     = lane & 15;
    const int khalf = (lane >> 4) * 2;   // 0 or 2

    const float* arow  = x + (size_t)(row0 + m) * D_FEAT;
    const float* b0row = W + (size_t)(col0 + m) * D_FEAT;        // tile 0: out-col = col0 + n
    const float* b1row = W + (size_t)(col0 + 16 + m) * D_FEAT;   // tile 1

    v8f c0 = {0.f, 0.f, 0.f, 0.f, 0.f, 0.f, 0.f, 0.f};
    v8f c1 = c0;

#pragma unroll 4
    for (int k = 0; k < D_FEAT; k += 4) {
        v2f a  = *(const v2f*)(arow  + k + khalf);
        v2f b0 = *(const v2f*)(b0row + k + khalf);
        v2f b1 = *(const v2f*)(b1row + k + khalf);
        c0 = __builtin_amdgcn_wmma_f32_16x16x4_f32(false, a, false, b0,
                                                   (short)0, c0, false, false);
        c1 = __builtin_amdgcn_wmma_f32_16x16x4_f32(false, a, false, b1,
                                                   (short)0, c1, false, false);
    }

    const int nlo  = lane & 15;
    const int rofs = (lane >> 4) * 8;    // rows r (lo half) / r+8 (hi half)
    const float bc0 = bias[col0 + nlo];
    const float bc1 = bias[col0 + 16 + nlo];
#pragma unroll
    for (int r = 0; r < 8; ++r) {
        const int row = row0 + r + rofs;
        const float s = norm[row];
        h[(size_t)row * D_FEAT + col0 + nlo]      = s * (c0[r] + bc0);
        h[(size_t)row * D_FEAT + col0 + 16 + nlo] = s * (c1[r] + bc1);
    }
}

// ---------------------------------------------------------------------------
// Kernel 5: edge aggregation. 64 threads per edge, 4 floats per thread.
// Gather h_scaled[cols[e]] (L2-resident: 102 MB < 192 MB L2), atomic-add into
// agg[rows[e]].
// ---------------------------------------------------------------------------
__global__ __launch_bounds__(256) void scatter_kernel(
    const int* __restrict__ rows, const int* __restrict__ cols,
    const float* __restrict__ h, float* __restrict__ agg, int n_edges) {
    const unsigned gid = blockIdx.x * blockDim.x + threadIdx.x;
    const unsigned e = gid >> 6;          // 64 threads per edge
    if (e >= (unsigned)n_edges) return;
    const int f = (gid & 63) * 4;         // feature offset 0..252
    const int r = rows[e];
    const int c = cols[e];
    const float4 v = *(const float4*)(h + (size_t)c * D_FEAT + f);
    float* dst = agg + (size_t)r * D_FEAT + f;
    atomicAdd(dst + 0, v.x);
    atomicAdd(dst + 1, v.y);
    atomicAdd(dst + 2, v.z);
    atomicAdd(dst + 3, v.w);
}

// ---------------------------------------------------------------------------
// Kernel 6: out = norm * (agg + h_scaled), in place over d_out
// ---------------------------------------------------------------------------
__global__ void final_kernel(float* __restrict__ out, const float* __restrict__ agg,
                             const float* __restrict__ norm, int n_elem) {
    int gid = blockIdx.x * blockDim.x + threadIdx.x;
    if (gid < n_elem) {
        const int row = gid >> 8;  // / 256
        out[gid] = norm[row] * (agg[gid] + out[gid]);
    }
}

// ---------------------------------------------------------------------------
extern "C" void kernel_launch(void* const* d_in, const int* in_sizes, int n_in,
                              void* d_out, int out_size, void* d_ws, size_t ws_size,
                              hipStream_t stream) {
    const float* x    = (const float*)d_in[0];
    const float* W    = (const float*)d_in[1];
    const float* bias = (const float*)d_in[2];
    const int*   rows = (const int*)d_in[3];
    const int*   cols = (const int*)d_in[4];

    const int n_nodes = in_sizes[0] / D_FEAT;
    const int n_edges = in_sizes[3];
    const int n_elem  = n_nodes * D_FEAT;

    float* h_scaled = (float*)d_out;                  // N*256, gather source
    float* agg      = (float*)d_ws;                   // N*256 accumulator
    float* nd       = agg + (size_t)n_elem;           // N floats: deg -> norm

    // 1. zero agg + deg
    {
        const int n4 = n_elem / 4;
        const int threads = 256;
        const int blocks = (n4 + threads - 1) / threads;
        zero_kernel<<<blocks, threads, 0, stream>>>((float4*)agg, nd, n4, n_nodes);
    }
    // 2. degree
    {
        const int threads = 256;
        const int blocks = (n_edges + threads - 1) / threads;
        deg_kernel<<<blocks, threads, 0, stream>>>(rows, nd, n_edges);
    }
    // 3. norm = rsqrt(1+deg)
    {
        const int threads = 256;
        const int blocks = (n_nodes + threads - 1) / threads;
        norm_kernel<<<blocks, threads, 0, stream>>>(nd, n_nodes);
    }
    // 4. h_scaled = norm * (x@W^T + bias)  [WMMA]
    {
        const int blocks = (n_nodes + 15) / 16;
        gemm_norm_kernel<<<blocks, 256, 0, stream>>>(x, W, bias, nd, h_scaled, n_nodes);
    }
    // 5. agg += gather/scatter over edges
    {
        const long long total = (long long)n_edges * 64;
        const int threads = 256;
        const int blocks = (int)((total + threads - 1) / threads);
        scatter_kernel<<<blocks, threads, 0, stream>>>(rows, cols, h_scaled, agg, n_edges);
    }
    // 6. out = norm * (agg + h_scaled), in place
    {
        const int threads = 256;
        const int blocks = (n_elem + threads - 1) / threads;
        final_kernel<<<blocks, threads, 0, stream>>>(h_scaled, agg, nd, n_elem);
    }
}